// MMCL_INV_62182536511797
// MI455X (gfx1250) — compile-verified
//
#include <hip/hip_runtime.h>
#include <math.h>

// ---------------------------------------------------------------------------
// MMCL contrastive loss for MI455X (gfx1250).
// Heavy part: 512 independent 511x511 f32 linear solves -> blocked LU with
// trailing update on V_WMMA_F32_16X16X4_F32 (keeps f32 precision).
// ---------------------------------------------------------------------------

#define D_FEAT 128
#define BS     512
#define NN     511          // 511 unknowns per system
#define NP     512          // padded system size (multiple of 32)
#define NB     32           // LU block size
#define SLOTS  128          // persistent solver workgroups (1 MB scratch each)
#define SIGMA_F 0.07f
#define REG_F   0.1f

typedef __attribute__((ext_vector_type(2))) float v2f;
typedef __attribute__((ext_vector_type(8))) float v8f;

// ---------------------------------------------------------------------------
// Kernel A: gather rows by index and L2-normalize (double-normalize in the
// reference is idempotent). One block of 128 threads per output row.
// ---------------------------------------------------------------------------
__global__ void mmcl_gather_norm(const float* __restrict__ x1,
                                 const float* __restrict__ x2,
                                 const int* __restrict__ i1,
                                 const int* __restrict__ i2,
                                 float* __restrict__ z) {
  __shared__ float red[128];
  int r = blockIdx.x;           // 0..1023
  int t = threadIdx.x;          // 0..127
  const float* src = (r < BS) ? (x1 + (size_t)i1[r] * D_FEAT)
                              : (x2 + (size_t)i2[r - BS] * D_FEAT);
  float v = src[t];
  red[t] = v * v;
  __syncthreads();
  for (int s = 64; s > 0; s >>= 1) {
    if (t < s) red[t] += red[t + s];
    __syncthreads();
  }
  float inv = 1.0f / fmaxf(sqrtf(red[0]), 1e-12f);
  z[(size_t)r * D_FEAT + t] = v * inv;
}

// ---------------------------------------------------------------------------
// Kernel B: K[i][j] = exp((z_i . z_j - 1)/sigma), i in [0,512), j in [0,1024).
// WMMA f32 16x16x4, one 16x16 tile per wave, K-loop over 128 features.
// A-frag layout (ISA 16x4 f32): lanes 0-15 hold K=0,1; lanes 16-31 hold K=2,3
// -> contiguous float2 per lane. B-frag (4x16) mirrors it.
// ---------------------------------------------------------------------------
__global__ void __launch_bounds__(256) mmcl_kgemm(const float* __restrict__ z,
                                                  float* __restrict__ Kmat) {
  int tid  = threadIdx.x;
  int lane = tid & 31;
  int wave = tid >> 5;
  int tile = blockIdx.x * 8 + wave;       // 0..2047  (32 x 64 tiles)
  int i0 = (tile >> 6) << 4;              // row tile * 16
  int j0 = (tile & 63) << 4;              // col tile * 16
  int half = lane >> 4;                   // 0: lanes 0-15, 1: lanes 16-31
  int l15  = lane & 15;

  const float* arow = z + (size_t)(i0 + l15) * D_FEAT + 2 * half;
  const float* brow = z + (size_t)(j0 + l15) * D_FEAT + 2 * half;

  v8f acc = {};
  for (int k0 = 0; k0 < D_FEAT; k0 += 4) {
    v2f a = *(const v2f*)(arow + k0);     // A[m][k0 + 2*half .. +1]
    v2f b = *(const v2f*)(brow + k0);     // B[k0 + 2*half .. +1][n]
    acc = __builtin_amdgcn_wmma_f32_16x16x4_f32(false, a, false, b,
                                                (short)0, acc, false, false);
  }
  int col = j0 + l15;
  int rbase = i0 + 8 * half;
#pragma unroll
  for (int r = 0; r < 8; ++r) {
    float d = acc[r];                     // dot(z_{rbase+r}, z_col)
    Kmat[(size_t)(rbase + r) * 1024 + col] = __expf((d - 1.0f) / SIGMA_F);
  }
}

// ---------------------------------------------------------------------------
// Kernel C: per system a: build DD (padded to 512), blocked no-pivot LU with
// WMMA trailing update, triangular solves, clip, partial-loss accumulation.
// ---------------------------------------------------------------------------
__global__ void __launch_bounds__(256) mmcl_solver(const float* __restrict__ Kmat,
                                                   float* __restrict__ Mslots,
                                                   float* __restrict__ partials) {
  extern __shared__ float smem[];
  float* kkA  = smem;                 //  512 : K row a
  float* bvec = smem + 512;           //  512 : rhs / solution
  float* Dblk = smem + 1024;          // 1024 : 32x32 diag block (later reduce scratch)
  float* Lp   = smem + 2048;          // 512*32 : -L panel, row-major stride NB
  float* Up   = Lp + NP * NB;         // 32*512 : U panel, row-major stride NP

  int tid  = threadIdx.x;
  int lane = tid & 31;
  int wave = tid >> 5;
  float* M = Mslots + (size_t)blockIdx.x * NP * NP;

  for (int a = blockIdx.x; a < BS; a += SLOTS) {
    // ---- stage K row a and rhs ------------------------------------------
    for (int i = tid; i < BS; i += 256) kkA[i] = Kmat[(size_t)a * 1024 + i];
    for (int i = tid; i < NP; i += 256) bvec[i] = (i < NN) ? 2.0f : 0.0f;
    __syncthreads();

    // ---- build DD[a] (padded with identity row/col 511) -----------------
    for (int e = tid; e < NP * NP; e += 256) {
      int r = e >> 9, c = e & (NP - 1);
      float v;
      if (r == NN || c == NN) {
        v = (r == c) ? 1.0f : 0.0f;
      } else {
        int ip = r + (r >= a);          // idx[a][r]
        int iq = c + (c >= a);          // idx[a][c]
        v = 1.0f + Kmat[(size_t)ip * 1024 + iq] - kkA[ip] - kkA[iq];
        if (r == c) v += REG_F;
      }
      M[e] = v;
    }
    __threadfence_block();
    __syncthreads();

    // ---- blocked LU, no pivoting ----------------------------------------
    for (int kb = 0; kb < NP / NB; ++kb) {
      int k0  = kb * NB;
      int o   = k0 + NB;
      int rem = NP - o;

      // load diagonal block into LDS
      for (int e = tid; e < NB * NB; e += 256)
        Dblk[e] = M[(size_t)(k0 + (e >> 5)) * NP + k0 + (e & 31)];
      __syncthreads();

      // factor 32x32 block in place (L unit-lower, U upper)
      for (int j = 0; j < NB - 1; ++j) {
        if (tid > j && tid < NB) Dblk[tid * NB + j] /= Dblk[j * NB + j];
        __syncthreads();
        for (int i = j + 1 + (tid >> 5); i < NB; i += 8) {
          int c = tid & 31;
          if (c > j) Dblk[i * NB + c] -= Dblk[i * NB + j] * Dblk[j * NB + c];
        }
        __syncthreads();
      }
      for (int e = tid; e < NB * NB; e += 256)
        M[(size_t)(k0 + (e >> 5)) * NP + k0 + (e & 31)] = Dblk[e];

      if (rem > 0) {
        // U panel: Up = L11^{-1} * A12   (one column per thread-iteration)
        for (int c = tid; c < rem; c += 256) {
          float col[NB];
#pragma unroll
          for (int r = 0; r < NB; ++r) col[r] = M[(size_t)(k0 + r) * NP + o + c];
#pragma unroll
          for (int r = 1; r < NB; ++r) {
            float s = col[r];
#pragma unroll
            for (int t2 = 0; t2 < r; ++t2) s -= Dblk[r * NB + t2] * col[t2];
            col[r] = s;
          }
#pragma unroll
          for (int r = 0; r < NB; ++r) {
            M[(size_t)(k0 + r) * NP + o + c] = col[r];
            Up[r * NP + c] = col[r];
          }
        }
        // L panel: L21 = A21 * U11^{-1}  (one row per thread-iteration),
        // stored NEGATED in LDS (f32 WMMA has no A-negate modifier).
        for (int rr = tid; rr < rem; rr += 256) {
          int ig = o + rr;
          float row[NB];
#pragma unroll
          for (int c = 0; c < NB; ++c) row[c] = M[(size_t)ig * NP + k0 + c];
#pragma unroll
          for (int c = 0; c < NB; ++c) {
            float s = row[c];
#pragma unroll
            for (int t2 = 0; t2 < c; ++t2) s -= row[t2] * Dblk[t2 * NB + c];
            row[c] = s / Dblk[c * NB + c];
          }
#pragma unroll
          for (int c = 0; c < NB; ++c) {
            M[(size_t)ig * NP + k0 + c] = row[c];
            Lp[rr * NB + c] = -row[c];
          }
        }
        __threadfence_block();
        __syncthreads();

        // trailing update: C += (-L) * U  via v_wmma_f32_16x16x4_f32.
        // 16x16 C tiles distributed over the block's 8 waves.
        int nT = rem >> 4;
        int half = lane >> 4;
        int l15  = lane & 15;
        for (int t = wave; t < nT * nT; t += 8) {
          int I0 = o + ((t / nT) << 4);
          int J0 = o + ((t % nT) << 4);
          v8f acc;
#pragma unroll
          for (int r = 0; r < 8; ++r)
            acc[r] = M[(size_t)(I0 + r + 8 * half) * NP + J0 + l15];
          const float* lpp = Lp + (size_t)(I0 - o + l15) * NB + 2 * half;
          const float* upp = Up + (J0 - o + l15);
#pragma unroll
          for (int kk = 0; kk < NB; kk += 4) {
            v2f af = *(const v2f*)(lpp + kk);          // contiguous float2
            v2f bf;
            bf.x = upp[(size_t)(kk + 2 * half)     * NP];
            bf.y = upp[(size_t)(kk + 2 * half + 1) * NP];
            acc = __builtin_amdgcn_wmma_f32_16x16x4_f32(false, af, false, bf,
                                                        (short)0, acc, false, false);
          }
#pragma unroll
          for (int r = 0; r < 8; ++r)
            M[(size_t)(I0 + r + 8 * half) * NP + J0 + l15] = acc[r];
        }
      }
      __threadfence_block();
      __syncthreads();
    }

    // ---- forward substitution: L y = b (unit diagonal) ------------------
    for (int j = 0; j < NP; ++j) {
      float bj = bvec[j];
      for (int i = j + 1 + tid; i < NP; i += 256)
        bvec[i] -= M[(size_t)i * NP + j] * bj;
      __syncthreads();
    }
    // ---- backward substitution: U x = y ---------------------------------
    for (int j = NP - 1; j >= 0; --j) {
      float xj = bvec[j] / M[(size_t)j * NP + j];
      __syncthreads();
      if (tid == 0) bvec[j] = xj;
      for (int i = tid; i < j; i += 256)
        bvec[i] -= M[(size_t)i * NP + j] * xj;
      __syncthreads();
    }

    // ---- partial loss: sum_p alpha_p * (Ks[ip][a] - Ks[a][a]) -----------
    float acc = 0.0f;
    float kdd = Kmat[(size_t)a * 1024 + BS + a];
    for (int p = tid; p < NN; p += 256) {
      float al = fminf(fmaxf(bvec[p], 0.0f), 1.0f);   // clip(relu(x),0,1)
      int ip = p + (p >= a);
      acc += al * (Kmat[(size_t)ip * 1024 + BS + a] - kdd);
    }
    Dblk[tid] = acc;                    // reuse Dblk as 256-wide scratch
    __syncthreads();
    for (int s = 128; s > 0; s >>= 1) {
      if (tid < s) Dblk[tid] += Dblk[tid + s];
      __syncthreads();
    }
    if (tid == 0) partials[a] = Dblk[0];
    __syncthreads();
  }
}

// ---------------------------------------------------------------------------
// Kernel D: deterministic fixed-order reduction -> scalar loss.
// ---------------------------------------------------------------------------
__global__ void mmcl_reduce(const float* __restrict__ partials,
                            float* __restrict__ out) {
  __shared__ float red[256];
  int t = threadIdx.x;
  red[t] = partials[t] + partials[t + 256];
  __syncthreads();
  for (int s = 128; s > 0; s >>= 1) {
    if (t < s) red[t] += red[t + s];
    __syncthreads();
  }
  if (t == 0) out[0] = red[0] * (1.0f / BS);
}

// ---------------------------------------------------------------------------
extern "C" void kernel_launch(void* const* d_in, const int* in_sizes, int n_in,
                              void* d_out, int out_size, void* d_ws, size_t ws_size,
                              hipStream_t stream) {
  (void)in_sizes; (void)n_in; (void)out_size; (void)ws_size;
  const float* x1 = (const float*)d_in[0];
  const float* x2 = (const float*)d_in[1];
  const int*   i1 = (const int*)d_in[2];
  const int*   i2 = (const int*)d_in[3];

  // workspace layout (floats): z | K | partials | M slots
  float* z        = (float*)d_ws;                       // 1024*128
  float* Kmat     = z + 1024 * D_FEAT;                  // 512*1024
  float* partials = Kmat + (size_t)BS * 1024;           // 512
  float* Mslots   = partials + BS;                      // SLOTS * 512*512

  size_t lds = (size_t)(2048 + NP * NB + NB * NP) * sizeof(float); // 136 KB
  hipFuncSetAttribute((const void*)mmcl_solver,
                      hipFuncAttributeMaxDynamicSharedMemorySize, (int)lds);

  mmcl_gather_norm<<<1024, 128, 0, stream>>>(x1, x2, i1, i2, z);
  mmcl_kgemm<<<256, 256, 0, stream>>>(z, Kmat);
  mmcl_solver<<<SLOTS, 256, lds, stream>>>(Kmat, Mslots, partials);
  mmcl_reduce<<<1, 256, 0, stream>>>(partials, (float*)d_out);
}